// LaplacianLoss_9431748182106
// MI455X (gfx1250) — compile-verified
//
#include <hip/hip_runtime.h>
#include <hip/hip_bf16.h>
#include <math.h>

// LaplacianLoss on MI455X (gfx1250):
//   Lx = L(16384x16384, fp32) @ verts(16384x3), loss = mean ||Lx_i||.
// Memory-bound (1.07GB of L @ 23.3 TB/s ~= 46us floor). Stream L through
// V_WMMA_F32_16X16X4_F32: A = 16x4 tile of L, B = 4x16 tile of verts^T.
// Only B/D columns 0..2 are meaningful; lanes n>=3 alias column 2 so all
// loads are unconditional and in-bounds (no EXEC churn in the hot loop).
// fp32 end-to-end; deterministic two-pass reduction (no float atomics).

typedef __attribute__((ext_vector_type(2))) float v2f;
typedef __attribute__((ext_vector_type(8))) float v8f;

#define NVERTS 16384
#define NK     16384
#define WAVES_PER_WG 8
#define KSPAN (NK / WAVES_PER_WG)   // 2048 per wave
#define NBLK  (NVERTS / 16)         // 1024 workgroups, one 16-row tile each

__global__ __launch_bounds__(256) void lap_tile_kernel(
    const float* __restrict__ L,
    const float* __restrict__ verts,
    float* __restrict__ wg_partials)
{
  const int tid  = threadIdx.x;
  const int lane = tid & 31;          // wave32
  const int wave = tid >> 5;          // 0..7 -> K split
  const int half = lane >> 4;         // 0: K+{0,1}, 1: K+{2,3}   (A/B layout)
  const int n    = lane & 15;         // A: row within tile; B/D: column

  const int m0  = blockIdx.x * 16;
  const int row = m0 + n;

  const int kBeg = wave * KSPAN;
  const int kEnd = kBeg + KSPAN;

  // A fragment base for this lane: L[row][k + 2*half .. +1]
  const float* Arow = L + (size_t)row * NK + 2 * half;

  // B fragment: b.x = verts[(k+2*half)*3 + nEff], b.y = same +1 vertex.
  // Lanes n>=3 produce garbage columns of D that are never read; aliasing
  // them to column 2 keeps every load in-bounds (max index = 16384*3-1)
  // with zero per-iteration predication.
  const int nEff = (n < 3) ? n : 2;
  const float* vBase = verts + (size_t)(2 * half) * 3 + nEff;

  v8f c = {0.f, 0.f, 0.f, 0.f, 0.f, 0.f, 0.f, 0.f};

  const int kPfMax = kEnd - 32;       // clamp keeps prefetch inside this row

  for (int k = kBeg; k < kEnd; k += 32) {
    // Branchless look-ahead: global_prefetch_b8 2KB ahead, clamped.
    const int kpf = (k + 512 < kPfMax) ? (k + 512) : kPfMax;
    __builtin_prefetch(Arow + kpf, 0, 3);
#pragma unroll
    for (int kk = 0; kk < 32; kk += 4) {
      const int kc = k + kk;
      // A: 8-byte aligned pair of fp32 from this lane's row
      v2f a = *(const v2f*)(Arow + kc);
      // B: unconditional, always in-bounds
      const float* vp = vBase + (size_t)kc * 3;
      v2f b;
      b.x = vp[0];
      b.y = vp[3];
      // D(16x16,f32) += A(16x4,f32) x B(4x16,f32)
      c = __builtin_amdgcn_wmma_f32_16x16x4_f32(
          /*neg_a=*/false, a, /*neg_b=*/false, b,
          /*c_mod=*/(short)0, c, /*reuse_a=*/false, /*reuse_b=*/false);
    }
  }

  // ---- reduce the 8 waves' accumulator fragments ----
  // D layout: VGPR j, lanes 0-15 -> (M=j, N=lane); lanes 16-31 -> (M=j+8, N=lane-16)
  __shared__ float red[WAVES_PER_WG][16][3];
  __shared__ float nrm[16];

  if (n < 3) {
#pragma unroll
    for (int j = 0; j < 8; ++j)
      red[wave][8 * half + j][n] = c[j];
  }
  __syncthreads();

  if (tid < 48) {                       // 16 rows x 3 components
    const int m = tid / 3, nn = tid % 3;
    float s = 0.f;
#pragma unroll
    for (int w = 0; w < WAVES_PER_WG; ++w) s += red[w][m][nn];
    red[0][m][nn] = s;                  // each (m,nn) owned by one thread
  }
  __syncthreads();

  if (tid < 16) {
    const float x = red[0][tid][0];
    const float y = red[0][tid][1];
    const float z = red[0][tid][2];
    nrm[tid] = sqrtf(x * x + y * y + z * z);
  }
  __syncthreads();

  if (tid == 0) {
    float s = 0.f;
#pragma unroll
    for (int i = 0; i < 16; ++i) s += nrm[i];
    wg_partials[blockIdx.x] = s;        // deterministic: fixed order, no atomics
  }
}

__global__ __launch_bounds__(256) void lap_reduce_kernel(
    const float* __restrict__ wg_partials,
    float* __restrict__ out)
{
  __shared__ float s[256];
  float v = 0.f;
  for (int i = threadIdx.x; i < NBLK; i += 256) v += wg_partials[i];
  s[threadIdx.x] = v;
  __syncthreads();
  for (int off = 128; off > 0; off >>= 1) {
    if (threadIdx.x < off) s[threadIdx.x] += s[threadIdx.x + off];
    __syncthreads();
  }
  if (threadIdx.x == 0) out[0] = s[0] * (1.0f / (float)NVERTS);
}

extern "C" void kernel_launch(void* const* d_in, const int* in_sizes, int n_in,
                              void* d_out, int out_size, void* d_ws, size_t ws_size,
                              hipStream_t stream) {
  const float* L     = (const float*)d_in[0];   // [16384,16384] fp32
  const float* verts = (const float*)d_in[1];   // [16384,3] fp32
  float* out = (float*)d_out;                   // scalar loss
  float* ws  = (float*)d_ws;                    // >= NBLK floats of scratch

  lap_tile_kernel<<<NBLK, 256, 0, stream>>>(L, verts, ws);
  lap_reduce_kernel<<<1, 256, 0, stream>>>(ws, out);
}